// RobustTripletLoss_28346784154016
// MI455X (gfx1250) — compile-verified
//
#include <hip/hip_runtime.h>
#include <hip/hip_bf16.h>

typedef __attribute__((ext_vector_type(2))) float v2f;
typedef __attribute__((ext_vector_type(8))) float v8f;

#define B 512
#define D 256
#define MARGIN 0.2f
#define EPS 1e-16f

// ---------------------------------------------------------------------------
// Kernel 1: sq[i] = dot(E[i], E[i]).  One wave (32 lanes) per row.
// ---------------------------------------------------------------------------
__global__ __launch_bounds__(256)
void row_sq_kernel(const float* __restrict__ E, float* __restrict__ sq) {
    const int lane   = threadIdx.x & 31;
    const int waveIn = threadIdx.x >> 5;              // 0..7
    const int row    = blockIdx.x * 8 + waveIn;       // 64 blocks * 8 waves = 512 rows
    const float* r = E + row * D;
    float acc = 0.0f;
#pragma unroll
    for (int u = 0; u < D / 32; ++u) {
        float v = r[lane + 32 * u];
        acc += v * v;
    }
#pragma unroll
    for (int off = 16; off > 0; off >>= 1)
        acc += __shfl_xor(acc, off, 32);
    if (lane == 0) sq[row] = acc;
}

// ---------------------------------------------------------------------------
// Kernel 2: 32x32 macro-tile of dist = sqrt(max(sq_i + sq_j - 2*G, EPS)).
// G accumulated with V_WMMA_F32_16X16X4_F32 over K=256, 2x2 register tiling:
// per k-step 2 A-frags + 2 B-frags (4 x b64 loads) feed 4 WMMAs
// -> 1.0 loads/WMMA (vs 2.0 untiled).
// A frag (16x4 f32): lane<16 -> K=k..k+1 of row i0+(lane&15);
//                    lane>=16 -> K=k+2..k+3.   B frag identical on j0 rows.
// ---------------------------------------------------------------------------
__device__ __forceinline__ v8f wmma_f32(v2f a, v2f b, v8f c) {
    return __builtin_amdgcn_wmma_f32_16x16x4_f32(
        /*neg_a=*/false, a, /*neg_b=*/false, b,
        /*c_mod=*/(short)0, c, /*reuse_a=*/false, /*reuse_b=*/false);
}

__global__ __launch_bounds__(32)
void dist_wmma_kernel(const float* __restrict__ E,
                      const float* __restrict__ sq,
                      float* __restrict__ dist) {
    const int j0   = blockIdx.x * 32;
    const int i0   = blockIdx.y * 32;
    const int lane = threadIdx.x;
    const int half = lane >> 4;       // 0 or 1
    const int t    = lane & 15;

    const float* a0 = E + (i0 + t) * D + 2 * half;
    const float* a1 = a0 + 16 * D;
    const float* b0 = E + (j0 + t) * D + 2 * half;
    const float* b1 = b0 + 16 * D;

    v8f c00 = {}, c01 = {}, c10 = {}, c11 = {};
#pragma unroll 4
    for (int k = 0; k < D; k += 4) {
        v2f A0 = *(const v2f*)(a0 + k);
        v2f A1 = *(const v2f*)(a1 + k);
        v2f B0 = *(const v2f*)(b0 + k);
        v2f B1 = *(const v2f*)(b1 + k);
        c00 = wmma_f32(A0, B0, c00);
        c01 = wmma_f32(A0, B1, c01);
        c10 = wmma_f32(A1, B0, c10);
        c11 = wmma_f32(A1, B1, c11);
    }

    // C/D layout: lane -> N = (lane&15); VGPR r -> M = r + 8*(lane>=16)
    const int mofs = half * 8;
#pragma unroll
    for (int q = 0; q < 4; ++q) {
        const v8f&  c  = (q == 0) ? c00 : (q == 1) ? c01 : (q == 2) ? c10 : c11;
        const int   ib = i0 + ((q >> 1) ? 16 : 0);
        const int   jb = j0 + ((q & 1) ? 16 : 0);
        const int   n  = jb + t;
        const float sqn = sq[n];
#pragma unroll
        for (int r = 0; r < 8; ++r) {
            const int i  = ib + mofs + r;
            float     d2 = sq[i] + sqn - 2.0f * c[r];
            d2 = fmaxf(d2, EPS);
            dist[i * B + n] = sqrtf(d2);
        }
    }
}

// ---------------------------------------------------------------------------
// Kernel 3: per-anchor batch-all triplet sum.  One block (256 thr) per anchor.
// Deterministic: fixed per-thread loop order, LDS tree reduction.
// ---------------------------------------------------------------------------
__global__ __launch_bounds__(256)
void anchor_kernel(const float* __restrict__ dist,
                   const int* __restrict__ labels,
                   float* __restrict__ avg,
                   int* __restrict__ valid) {
    __shared__ float drow[B];
    __shared__ int   lab[B];
    __shared__ float reds[256];
    __shared__ int   redp[256];
    __shared__ int   redn[256];

    const int a   = blockIdx.x;
    const int tid = threadIdx.x;

    for (int idx = tid; idx < B; idx += 256) {
        drow[idx] = dist[a * B + idx];
        lab[idx]  = labels[idx];
    }
    __syncthreads();

    const int la = lab[a];
    float s     = 0.0f;
    int   nposL = 0;
    int   nnegL = 0;

    for (int p = tid; p < B; p += 256) {
        const bool isNeg = (lab[p] != la);
        const bool isPos = !isNeg && (p != a);
        nnegL += isNeg ? 1 : 0;
        if (isPos) {
            ++nposL;
            const float dp = drow[p];
            for (int n = 0; n < B; ++n) {
                const float l = fmaxf(dp - drow[n] + MARGIN, 0.0f);
                s += (lab[n] != la) ? l : 0.0f;
            }
        }
    }

    reds[tid] = s; redp[tid] = nposL; redn[tid] = nnegL;
    __syncthreads();
    for (int off = 128; off > 0; off >>= 1) {
        if (tid < off) {
            reds[tid] += reds[tid + off];
            redp[tid] += redp[tid + off];
            redn[tid] += redn[tid + off];
        }
        __syncthreads();
    }
    if (tid == 0) {
        const long long cnt = (long long)redp[0] * (long long)redn[0];
        avg[a]   = (cnt > 0) ? reds[0] / (float)cnt : 0.0f;
        valid[a] = (cnt > 0) ? 1 : 0;
    }
}

// ---------------------------------------------------------------------------
// Kernel 4: final deterministic reduction over anchors.
// out[0] = loss, out[1] = (float)num_valid
// ---------------------------------------------------------------------------
__global__ __launch_bounds__(256)
void finalize_kernel(const float* __restrict__ avg,
                     const int* __restrict__ valid,
                     float* __restrict__ out) {
    __shared__ float reds[256];
    __shared__ int   redv[256];
    const int tid = threadIdx.x;
    reds[tid] = avg[tid] + avg[tid + 256];
    redv[tid] = valid[tid] + valid[tid + 256];
    __syncthreads();
    for (int off = 128; off > 0; off >>= 1) {
        if (tid < off) {
            reds[tid] += reds[tid + off];
            redv[tid] += redv[tid + off];
        }
        __syncthreads();
    }
    if (tid == 0) {
        const int nv = redv[0];
        out[0] = (nv > 0) ? reds[0] / (float)nv : 0.0f;
        out[1] = (float)nv;
    }
}

// ---------------------------------------------------------------------------
extern "C" void kernel_launch(void* const* d_in, const int* in_sizes, int n_in,
                              void* d_out, int out_size, void* d_ws, size_t ws_size,
                              hipStream_t stream) {
    const float* E      = (const float*)d_in[0];   // [512, 256] f32
    const int*   labels = (const int*)d_in[1];     // [512] int
    float*       out    = (float*)d_out;

    char* ws = (char*)d_ws;
    float* dist  = (float*)(ws);                                   // 512*512*4 = 1 MB
    float* sq    = (float*)(ws + (size_t)B * B * sizeof(float));   // 512 f
    float* avg   = (float*)(ws + (size_t)B * B * sizeof(float) + B * sizeof(float));
    int*   valid = (int*)  (ws + (size_t)B * B * sizeof(float) + 2 * B * sizeof(float));

    // 1) row norms: 512 waves = 64 blocks * 8 waves
    row_sq_kernel<<<64, 256, 0, stream>>>(E, sq);

    // 2) fused Gram + sqrt-distance, WMMA f32 16x16x4, 2x2-tiled, 1 wave / 32x32 tile
    dist_wmma_kernel<<<dim3(B / 32, B / 32), 32, 0, stream>>>(E, sq, dist);

    // 3) per-anchor triplet sums
    anchor_kernel<<<B, 256, 0, stream>>>(dist, labels, avg, valid);

    // 4) final reduction -> loss, num_valid
    finalize_kernel<<<1, 256, 0, stream>>>(avg, valid, out);
}